// LangevinApproxDRO_403726926610
// MI455X (gfx1250) — compile-verified
//
#include <hip/hip_runtime.h>
#include <math.h>

// ---------------------------------------------------------------------------
// LangevinApproxDRO for MI455X (gfx1250).
// Bandwidth-bound (z is 21 MB, L2-resident); residual GEMV done with
// V_WMMA_F32_16X16X4_F32 (full f32 precision, engages the matrix pipe).
// ---------------------------------------------------------------------------

#define B_SZ      4096
#define D_FEAT    128          // feature dims (w length)
#define D_FULL    129          // features + label
#define MP        10           // M_PARALLEL
#define T_STEPS   10
#define M_TRAIN   10
#define NROWS     (B_SZ * MP)  // 40960 Langevin chains
#define LR_F      0.1f
#define SQRT_2LR  0.4472135954999579f   // sqrt(2*LR), SIGMA_LANGEVIN = 1
#define BOUND_G   100.0f
#define RHO_F     1.0f
// EPS = SIGMA = 1  ->  (lbd + EPS/SIGMA^2) = lbd + 1

typedef float v2f __attribute__((ext_vector_type(2)));
typedef float v8f __attribute__((ext_vector_type(8)));

// ---------------- counter-based gaussian noise (PCG hash + Box-Muller) ------
__device__ __forceinline__ unsigned pcg_hash(unsigned v) {
    v = v * 747796405u + 2891336453u;
    unsigned x = ((v >> ((v >> 28u) + 4u)) ^ v) * 277803737u;
    return (x >> 22u) ^ x;
}

__device__ __forceinline__ float gauss_noise(unsigned idx, int t) {
    unsigned h1 = pcg_hash(idx ^ (0x9E3779B9u * (unsigned)(t + 1)));
    unsigned h2 = pcg_hash(h1 ^ 0x85EBCA6Bu);
    float u1 = (float)((h1 >> 8) + 1u) * (1.0f / 16777216.0f);  // (0,1]
    float u2 = (float)(h2 >> 8) * (1.0f / 16777216.0f);         // [0,1)
    return sqrtf(-2.0f * __logf(u1)) * __cosf(6.28318530718f * u2);
}

// ---------------- init kernels ---------------------------------------------
__global__ void k_zero(float* __restrict__ p, int n) {
    int i = blockIdx.x * blockDim.x + threadIdx.x;
    if (i < n) p[i] = 0.0f;
}

// z[row, d] = (d < 128) ? x[row/MP, d] : y[row/MP]
__global__ void k_init_z(float* __restrict__ z,
                         const float* __restrict__ x,
                         const float* __restrict__ y) {
    size_t e = (size_t)blockIdx.x * blockDim.x + threadIdx.x;
    if (e >= (size_t)NROWS * D_FULL) return;
    int row = (int)(e / D_FULL);
    int d   = (int)(e % D_FULL);
    int b   = row / MP;
    z[e] = (d < D_FEAT) ? x[(size_t)b * D_FEAT + d] : y[b];
}

// ---------------- pass 1: gradient + global ||g||^2 (WMMA residuals) -------
// One wave handles 16 rows. Residual pred = z[row,0:128] . w via 32 chained
// V_WMMA_F32_16X16X4_F32 with B = w-chunk replicated into all 16 columns,
// so every D column equals pred_m. Then lanes sweep the 16x129 gradient tile
// accumulating ||g||^2, reduced per wave and atomically added per step slot.
__global__ void __launch_bounds__(256)
k_grad_norm(const float* __restrict__ z,
            const float* __restrict__ x,
            const float* __restrict__ y,
            const float* __restrict__ w,
            const float* __restrict__ lbdp,
            float* __restrict__ normsq_out) {
    __shared__ float s_r[8 * 16];           // residual r per row, per wave
    const int lane    = threadIdx.x & 31;
    const int wv      = threadIdx.x >> 5;
    const int half    = lane >> 4;          // 0: lanes 0-15, 1: lanes 16-31
    const int l16     = lane & 15;
    const int rowBase = (blockIdx.x * 8 + wv) * 16;   // exact fit: no guard,
                                                      // EXEC all-1s for WMMA
    // A 16x4 f32 layout (ISA 7.12.2): lane l<16 -> M=l, VGPR j -> K=j;
    // lanes 16-31 -> K = j+2.  B 4x16: mirror K mapping, N = lane&15.
    const float* zr = z + (size_t)(rowBase + l16) * D_FULL;
    v8f acc = {0.f, 0.f, 0.f, 0.f, 0.f, 0.f, 0.f, 0.f};
    for (int k = 0; k < D_FEAT; k += 4) {
        const int k0 = k + 2 * half;
        v2f a, bm;
        a.x  = zr[k0 + 0];
        a.y  = zr[k0 + 1];
        bm.x = w[k0 + 0];
        bm.y = w[k0 + 1];
        // 8 args: (neg_a, A, neg_b, B, c_mod, C, reuse_a, reuse_b)
        acc = __builtin_amdgcn_wmma_f32_16x16x4_f32(
            false, a, false, bm, (short)0, acc, false, false);
    }
    // C/D layout: lane<16 holds M = vgpr (N=lane), lane>=16 holds M = vgpr+8.
    // All columns identical -> lanes 0 and 16 publish the 16 residuals.
    if (l16 == 0) {
#pragma unroll
        for (int i = 0; i < 8; ++i) {
            const int m = i + 8 * half;
            // r = pred - z[row, 128]
            s_r[wv * 16 + m] = acc[i] - z[(size_t)(rowBase + m) * D_FULL + D_FEAT];
        }
    }
    __syncthreads();

    const float lbd1 = lbdp[0] + 1.0f;      // lbd + EPS/SIGMA^2
    float nsq = 0.0f;
#pragma unroll 4
    for (int m = 0; m < 16; ++m) {
        const int row = rowBase + m;
        const int b   = row / MP;
        const float r = s_r[wv * 16 + m];
        const float* zrow = z + (size_t)row * D_FULL;
#pragma unroll
        for (int c = 0; c < 4; ++c) {
            const int d  = lane + 32 * c;
            const float zd = zrow[d];
            const float g  = 2.0f * r * w[d] - 2.0f * lbd1 * (zd - x[(size_t)b * D_FEAT + d]);
            nsq += g * g;
        }
        if (lane == 0) {
            const float zd = zrow[D_FEAT];
            const float g  = -2.0f * r - 2.0f * lbd1 * (zd - y[b]);
            nsq += g * g;
        }
    }
    for (int off = 16; off > 0; off >>= 1) nsq += __shfl_down(nsq, off, 32);
    if (lane == 0) atomicAdd(normsq_out, nsq);
}

// ---------------- pass 2: scale + noise update + sample-loss accumulation --
// One lane per chain row. Recomputes r from the (L2-resident) z, applies the
// clipped Langevin step, and since T == M_TRAIN every step's state is a
// sample: accumulate adv_loss - lbd*cost into the global loss accumulator.
__global__ void __launch_bounds__(256)
k_update(float* __restrict__ z,
         const float* __restrict__ x,
         const float* __restrict__ y,
         const float* __restrict__ w,
         const float* __restrict__ lbdp,
         const float* __restrict__ normsq,
         int t,
         float* __restrict__ loss_acc) {
    const int row = blockIdx.x * blockDim.x + threadIdx.x;  // exact fit (40960)
    const int b   = row / MP;
    const float lbd  = lbdp[0];
    const float lbd1 = lbd + 1.0f;
    const float nrm  = sqrtf(normsq[0]);
    const float s    = BOUND_G / fmaxf(BOUND_G, nrm);       // gradient clip
    float* zrow = z + (size_t)row * D_FULL;

    float pred = 0.0f;
    for (int d = 0; d < D_FEAT; ++d) pred += zrow[d] * w[d];
    const float r = pred - zrow[D_FEAT];

    const unsigned base = (unsigned)row * (unsigned)D_FULL;
    float pred_new = 0.0f, cost = 0.0f, rn = 0.0f;
    for (int d = 0; d <= D_FEAT; ++d) {
        const float zd  = zrow[d];
        const float xid = (d < D_FEAT) ? x[(size_t)b * D_FEAT + d] : y[b];
        const float g   = (d < D_FEAT)
                        ? (2.0f * r * w[d] - 2.0f * lbd1 * (zd - xid))
                        : (-2.0f * r - 2.0f * lbd1 * (zd - xid));
        const float zn = zd + LR_F * s * g + SQRT_2LR * gauss_noise(base + (unsigned)d, t);
        zrow[d] = zn;
        const float diff = zn - xid;
        cost += diff * diff;
        if (d < D_FEAT) pred_new += zn * w[d];
        else            rn = pred_new - zn;
    }
    float contrib = rn * rn - lbd * cost;   // adv_loss - lbd*cost
    for (int off = 16; off > 0; off >>= 1) contrib += __shfl_down(contrib, off, 32);
    if ((threadIdx.x & 31) == 0) atomicAdd(loss_acc, contrib);
}

// ---------------- finalize --------------------------------------------------
__global__ void k_final(const float* __restrict__ loss_acc,
                        const float* __restrict__ lbdp,
                        float* __restrict__ out) {
    const float denom = (float)NROWS * (float)M_TRAIN;      // 409600 samples
    out[0] = lbdp[0] * RHO_F + loss_acc[0] / denom;
}

// ---------------- host-side orchestration ----------------------------------
extern "C" void kernel_launch(void* const* d_in, const int* in_sizes, int n_in,
                              void* d_out, int out_size, void* d_ws, size_t ws_size,
                              hipStream_t stream) {
    const float* x   = (const float*)d_in[0];   // (4096, 128)
    const float* y   = (const float*)d_in[1];   // (4096,)
    const float* w   = (const float*)d_in[2];   // (128,)
    const float* lbd = (const float*)d_in[3];   // (1,)
    float* out = (float*)d_out;

    // workspace: z (NROWS*129 f32 = 21.1 MB) | norms[T] | loss[1]
    float* ws    = (float*)d_ws;
    float* z     = ws;
    float* norms = ws + (size_t)NROWS * D_FULL;
    float* loss  = norms + T_STEPS;

    k_zero<<<1, 32, 0, stream>>>(norms, T_STEPS + 1);

    const size_t ztot = (size_t)NROWS * D_FULL;
    k_init_z<<<(unsigned)((ztot + 255) / 256), 256, 0, stream>>>(z, x, y);

    for (int t = 0; t < T_STEPS; ++t) {
        // 8 waves/block * 16 rows/wave = 128 rows/block -> 320 blocks exactly
        k_grad_norm<<<NROWS / 128, 256, 0, stream>>>(z, x, y, w, lbd, norms + t);
        // lane-per-row update: 160 blocks exactly
        k_update<<<NROWS / 256, 256, 0, stream>>>(z, x, y, w, lbd, norms + t, t, loss);
    }

    k_final<<<1, 1, 0, stream>>>(loss, lbd, out);
}